// GNNModel_7206955123357
// MI455X (gfx1250) — compile-verified
//
#include <hip/hip_runtime.h>
#include <math.h>

typedef float v2f __attribute__((ext_vector_type(2)));
typedef float v8f __attribute__((ext_vector_type(8)));

#define GN_NODES 100000
#define GN_EDGES 1600000
#define GC_IN    64
#define GC_HID   64
#define GC_OUT   32

// ---------------------------------------------------------------------------
// degree / normalization precompute
// ---------------------------------------------------------------------------
__global__ void gcn_init_deg(float* __restrict__ deg, int n) {
    int i = blockIdx.x * blockDim.x + threadIdx.x;
    if (i < n) deg[i] = 1.0f;  // self-loop edge weight 1
}

__global__ void gcn_deg_accum(const int* __restrict__ ei, const float* __restrict__ ew,
                              float* __restrict__ deg, int ne) {
    int e = blockIdx.x * blockDim.x + threadIdx.x;
    if (e < ne) atomicAdd(&deg[ei[ne + e]], ew[e]);  // dst side
}

__global__ void gcn_dinv(const float* __restrict__ deg, float* __restrict__ dinv, int n) {
    int i = blockIdx.x * blockDim.x + threadIdx.x;
    if (i < n) {
        float d = deg[i];
        dinv[i] = (d > 0.0f) ? rsqrtf(d) : 0.0f;
    }
}

__global__ void gcn_edge_norm(const int* __restrict__ ei, const float* __restrict__ ew,
                              const float* __restrict__ dinv, float* __restrict__ norm, int ne) {
    int e = blockIdx.x * blockDim.x + threadIdx.x;
    if (e < ne) norm[e] = dinv[ei[e]] * ew[e] * dinv[ei[ne + e]];
}

// ---------------------------------------------------------------------------
// dense GEMM H[M x NCOLS] = A[M x 64] * W[64 x NCOLS] using fp32 WMMA.
// NCOLS is a compile-time constant so B loads become saddr + immediate-offset
// (no per-iteration 64-bit address math). One wave per 16x16 output tile;
// K stepped by 4 (V_WMMA_F32_16X16X4_F32), exact fp32 like the reference.
// ---------------------------------------------------------------------------
template <int NCOLS>
__global__ void gcn_gemm_wmma_f32(const float* __restrict__ A,
                                  const float* __restrict__ W,
                                  float* __restrict__ H,
                                  int m_tiles) {
    const int waves_per_block = blockDim.x >> 5;
    const int wave    = blockIdx.x * waves_per_block + (threadIdx.x >> 5);
    const int n_tiles = NCOLS >> 4;
    if (wave >= m_tiles * n_tiles) return;   // wave-uniform: EXEC all-1s inside

    const int mt   = wave / n_tiles;
    const int nt   = wave - mt * n_tiles;
    const int lane = threadIdx.x & 31;
    const int half = lane >> 4;   // 0: lanes 0-15, 1: lanes 16-31
    const int l16  = lane & 15;

    // A row base for this lane; per-iteration offset is a constant immediate.
    const float* arow = A + (mt * 16 + l16) * 64 + (half << 1);
    // B column base for this lane; per-iteration offset ka*NCOLS is constant.
    const float* wcol = W + (half << 1) * NCOLS + nt * 16 + l16;

    v8f acc = {};
#pragma unroll
    for (int kk = 0; kk < 64; kk += 4) {
        // A 16x4 fp32 layout: lane holds {A[m][ka], A[m][ka+1]}, ka = kk + 2*half
        v2f a = *(const v2f*)(arow + kk);
        // B 4x16 fp32 layout: VGPR0 = row ka, VGPR1 = row ka+1, striped over lanes
        v2f b;
        b.x = wcol[kk * NCOLS];
        b.y = wcol[kk * NCOLS + NCOLS];
        acc = __builtin_amdgcn_wmma_f32_16x16x4_f32(
            /*neg_a=*/false, a, /*neg_b=*/false, b,
            /*c_mod=*/(short)0, acc, /*reuse_a=*/false, /*reuse_b=*/false);
    }

    // C/D 16x16 fp32 layout: acc[r] -> row mt*16 + half*8 + r, col nt*16+l16
    float* hbase = H + (mt * 16 + (half << 3)) * NCOLS + nt * 16 + l16;
#pragma unroll
    for (int r = 0; r < 8; ++r) hbase[r * NCOLS] = acc[r];
}

// ---------------------------------------------------------------------------
// self-loop init:  agg[i][c] = h[i][c] * dinv[i]^2      (C = 64, float4 lanes)
// ---------------------------------------------------------------------------
__global__ void gcn_self_init64(const float* __restrict__ h, const float* __restrict__ dinv,
                                float* __restrict__ agg, int n) {
    int t = blockIdx.x * blockDim.x + threadIdx.x;
    if (t >= n * 16) return;
    int node = t >> 4, q = t & 15;
    float s = dinv[node]; s *= s;
    float4 v = *(const float4*)(h + node * 64 + q * 4);
    float4 o; o.x = v.x * s; o.y = v.y * s; o.z = v.z * s; o.w = v.w * s;
    *(float4*)(agg + node * 64 + q * 4) = o;
}

// edge scatter, C = 64: 16 lanes per edge, float4 gather + 4 atomics (L2-resident dst)
__global__ void gcn_scatter64(const int* __restrict__ ei, const float* __restrict__ norm,
                              const float* __restrict__ h, float* __restrict__ agg, int ne) {
    int t = blockIdx.x * blockDim.x + threadIdx.x;
    if (t >= ne * 16) return;
    int e = t >> 4, q = t & 15;
    int s = ei[e], d = ei[ne + e];
    float w = norm[e];
    float4 v = *(const float4*)(h + s * 64 + q * 4);
    float* o = agg + d * 64 + q * 4;
    atomicAdd(o + 0, v.x * w);
    atomicAdd(o + 1, v.y * w);
    atomicAdd(o + 2, v.z * w);
    atomicAdd(o + 3, v.w * w);
}

// in-place bias + ReLU on 64-channel aggregate
__global__ void gcn_bias_relu64(float* __restrict__ agg, const float* __restrict__ b, int n) {
    int t = blockIdx.x * blockDim.x + threadIdx.x;
    if (t >= n * 16) return;
    int node = t >> 4, q = t & 15;
    float4 bb = *(const float4*)(b + q * 4);
    float4 v  = *(float4*)(agg + node * 64 + q * 4);
    v.x = fmaxf(v.x + bb.x, 0.0f);
    v.y = fmaxf(v.y + bb.y, 0.0f);
    v.z = fmaxf(v.z + bb.z, 0.0f);
    v.w = fmaxf(v.w + bb.w, 0.0f);
    *(float4*)(agg + node * 64 + q * 4) = v;
}

// layer-2 output init: out[i][c] = h2[i][c]*dinv[i]^2 + b2[c]   (C = 32)
__global__ void gcn_out_init32(const float* __restrict__ h2, const float* __restrict__ dinv,
                               const float* __restrict__ b2, float* __restrict__ out, int n) {
    int t = blockIdx.x * blockDim.x + threadIdx.x;
    if (t >= n * 8) return;
    int node = t >> 3, q = t & 7;
    float s = dinv[node]; s *= s;
    float4 bb = *(const float4*)(b2 + q * 4);
    float4 v  = *(const float4*)(h2 + node * 32 + q * 4);
    float4 o; o.x = v.x * s + bb.x; o.y = v.y * s + bb.y;
    o.z = v.z * s + bb.z; o.w = v.w * s + bb.w;
    *(float4*)(out + node * 32 + q * 4) = o;
}

// edge scatter, C = 32: 8 lanes per edge
__global__ void gcn_scatter32(const int* __restrict__ ei, const float* __restrict__ norm,
                              const float* __restrict__ h2, float* __restrict__ out, int ne) {
    int t = blockIdx.x * blockDim.x + threadIdx.x;
    if (t >= ne * 8) return;
    int e = t >> 3, q = t & 7;
    int s = ei[e], d = ei[ne + e];
    float w = norm[e];
    float4 v = *(const float4*)(h2 + s * 32 + q * 4);
    float* o = out + d * 32 + q * 4;
    atomicAdd(o + 0, v.x * w);
    atomicAdd(o + 1, v.y * w);
    atomicAdd(o + 2, v.z * w);
    atomicAdd(o + 3, v.w * w);
}

// ---------------------------------------------------------------------------
extern "C" void kernel_launch(void* const* d_in, const int* in_sizes, int n_in,
                              void* d_out, int out_size, void* d_ws, size_t ws_size,
                              hipStream_t stream) {
    const float* x  = (const float*)d_in[0];
    const int*   ei = (const int*)d_in[1];   // [2, E]: row 0 = src, row 1 = dst
    const float* ew = (const float*)d_in[2];
    const float* W1 = (const float*)d_in[3];
    const float* b1 = (const float*)d_in[4];
    const float* W2 = (const float*)d_in[5];
    const float* b2 = (const float*)d_in[6];
    float* out = (float*)d_out;

    const int N = GN_NODES, E = GN_EDGES;

    // workspace layout (float offsets, all 16B-aligned)
    float* ws   = (float*)d_ws;
    float* deg  = ws;                 //   100000
    float* dinv = ws + 100000;        //   100000
    float* norm = ws + 200000;        //  1600000
    float* h    = ws + 1800000;       //  6400000 (h1; reused for h2: 3200000)
    float* agg  = ws + 8200000;       //  6400000 (layer-1 aggregate / z1)

    const int T = 256;

    // normalization precompute
    gcn_init_deg  <<<(N + T - 1) / T, T, 0, stream>>>(deg, N);
    gcn_deg_accum <<<(E + T - 1) / T, T, 0, stream>>>(ei, ew, deg, E);
    gcn_dinv      <<<(N + T - 1) / T, T, 0, stream>>>(deg, dinv, N);
    gcn_edge_norm <<<(E + T - 1) / T, T, 0, stream>>>(ei, ew, dinv, norm, E);

    // layer 1: h1 = x @ W1   (6250 x 4 = 25000 tiles, 8 waves/block)
    gcn_gemm_wmma_f32<GC_HID><<<(25000 + 7) / 8, T, 0, stream>>>(x, W1, h, N / 16);
    gcn_self_init64  <<<(N * 16 + T - 1) / T, T, 0, stream>>>(h, dinv, agg, N);
    gcn_scatter64    <<<(E * 16 + T - 1) / T, T, 0, stream>>>(ei, norm, h, agg, E);
    gcn_bias_relu64  <<<(N * 16 + T - 1) / T, T, 0, stream>>>(agg, b1, N);

    // layer 2: h2 = z1 @ W2  (6250 x 2 = 12500 tiles)
    gcn_gemm_wmma_f32<GC_OUT><<<(12500 + 7) / 8, T, 0, stream>>>(agg, W2, h, N / 16);
    gcn_out_init32   <<<(N * 8 + T - 1) / T, T, 0, stream>>>(h, dinv, b2, out, N);
    gcn_scatter32    <<<(E * 8 + T - 1) / T, T, 0, stream>>>(ei, norm, h, out, E);
}